// MultiConvolveNet_16492674417204
// MI455X (gfx1250) — compile-verified
//
#include <hip/hip_runtime.h>
#include <hip/hip_bf16.h>

// CDNA5 / gfx1250, wave32. WMMA fp32 16x16x4 for the two per-layer GEMMs;
// streaming wave-per-edge scatter with f32 atomics for message aggregation.

typedef float v2f __attribute__((ext_vector_type(2)));
typedef float v8f __attribute__((ext_vector_type(8)));

#define N_NODES 50000
#define N_EDGES 800000
#define DIM     128

// ---------------------------------------------------------------------------
// zero a float buffer (grid-stride)
// ---------------------------------------------------------------------------
__global__ void zero_f32(float* __restrict__ p, size_t n) {
  size_t i = (size_t)blockIdx.x * blockDim.x + threadIdx.x;
  size_t stride = (size_t)gridDim.x * blockDim.x;
  for (; i < n; i += stride) p[i] = 0.0f;
}

// ---------------------------------------------------------------------------
// n = relu(h @ Qw^T + Qb)          h:[N,128]  Qw:[128,128]  n:[N,128]
// block = 256 threads = 8 wave32; wave w owns the 16x16 tile at cols w*16,
// rows blockIdx.x*16. K-loop in steps of 4 using V_WMMA_F32_16X16X4_F32.
//
// fp32 A 16x4 layout: lanes 0-15 -> (K=k, K=k+1), lanes 16-31 -> (K=k+2,k+3),
// M = lane%16  => one 8-byte load per lane per step.
// B 4x16 is the mirrored layout; since we need B[k][j] = Qw[j][k] and Qw is
// row-major in j, the per-lane fragment is again 2 consecutive floats.
// ---------------------------------------------------------------------------
__global__ __launch_bounds__(256) void gemm_q_relu(
    const float* __restrict__ h, const float* __restrict__ Qw,
    const float* __restrict__ Qb, float* __restrict__ n_out) {
  const int wave = threadIdx.x >> 5;
  const int lane = threadIdx.x & 31;
  const int lr = lane & 15;       // 0..15
  const int hi = lane >> 4;       // 0 or 1
  const int row0 = blockIdx.x * 16;
  const int col0 = wave * 16;

  const float* arow = h + (size_t)(row0 + lr) * DIM + 2 * hi;
  const float* brow = Qw + (size_t)(col0 + lr) * DIM + 2 * hi;

  v8f c = {0.f, 0.f, 0.f, 0.f, 0.f, 0.f, 0.f, 0.f};
#pragma unroll 8
  for (int k = 0; k < DIM; k += 4) {
    v2f a = *(const v2f*)(arow + k);
    v2f b = *(const v2f*)(brow + k);
    c = __builtin_amdgcn_wmma_f32_16x16x4_f32(
        /*neg_a=*/false, a, /*neg_b=*/false, b,
        /*c_mod=*/(short)0, c, /*reuse_a=*/false, /*reuse_b=*/false);
  }

  // C/D layout: VGPR v, lanes 0-15 -> M=v, lanes 16-31 -> M=v+8; N = lane%16
  const int col = col0 + lr;
  const float bias = Qb[col];
#pragma unroll
  for (int v = 0; v < 8; ++v) {
    const int row = row0 + v + 8 * hi;
    float x = c[v] + bias;
    n_out[(size_t)row * DIM + col] = fmaxf(x, 0.0f);
  }
}

// ---------------------------------------------------------------------------
// Weighted scatter-sum: one wave32 per edge, 4 floats per lane (float4 gather,
// 4x global_atomic_add_f32 scatter); lane 0 adds w to wsum[dst].
// ---------------------------------------------------------------------------
__global__ __launch_bounds__(256) void edge_scatter(
    const float* __restrict__ nsrc, const float* __restrict__ ew,
    const int* __restrict__ src, const int* __restrict__ dst,
    float* __restrict__ agg, float* __restrict__ wsum) {
  const int e = (int)((blockIdx.x * (unsigned)blockDim.x + threadIdx.x) >> 5);
  const int lane = threadIdx.x & 31;
  if (e >= N_EDGES) return;

  // prefetch ahead on the index/weight streams (global_prefetch_b8)
  if (e + 256 < N_EDGES) {
    __builtin_prefetch(&src[e + 256], 0, 0);
    __builtin_prefetch(&dst[e + 256], 0, 0);
    __builtin_prefetch(&ew[e + 256], 0, 0);
  }

  const int s = src[e];
  const int d = dst[e];
  const float w = ew[e];

  const float4 v = *(const float4*)(nsrc + (size_t)s * DIM + lane * 4);
  float* ap = agg + (size_t)d * DIM + lane * 4;
  atomicAdd(ap + 0, v.x * w);
  atomicAdd(ap + 1, v.y * w);
  atomicAdd(ap + 2, v.z * w);
  atomicAdd(ap + 3, v.w * w);
  if (lane == 0) atomicAdd(&wsum[d], w);
}

// ---------------------------------------------------------------------------
// z = relu(concat([agg / clip(ws,1), h]) @ Ww^T + Wb); h_out = z / ||z||_2.
// Block = 16 rows x all 128 cols (8 waves, one 16x16 tile each) so the L2
// row-norm reduces inside the block via LDS ds_add_f32 atomics.
// Safe in place (hin may alias hout): each block only reads/writes its rows,
// and all reads precede writes (enforced by dataflow + barriers).
// ---------------------------------------------------------------------------
__global__ __launch_bounds__(256) void gemm_w_relu_norm(
    const float* __restrict__ agg, const float* __restrict__ ws,
    const float* hin, const float* __restrict__ Ww,
    const float* __restrict__ Wb, float* hout) {
  __shared__ float ssq[16];

  const int wave = threadIdx.x >> 5;
  const int lane = threadIdx.x & 31;
  const int lr = lane & 15;
  const int hi = lane >> 4;
  const int row0 = blockIdx.x * 16;
  const int col0 = wave * 16;
  const int arow = row0 + lr;

  const float scale = 1.0f / fmaxf(ws[arow], 1.0f);   // clip(segment_sum, 1, inf)
  const float* ap0 = agg + (size_t)arow * DIM + 2 * hi;
  const float* ap1 = hin + (size_t)arow * DIM + 2 * hi;
  const float* brow = Ww + (size_t)(col0 + lr) * (2 * DIM) + 2 * hi;

  v8f c = {0.f, 0.f, 0.f, 0.f, 0.f, 0.f, 0.f, 0.f};
#pragma unroll 8
  for (int k = 0; k < DIM; k += 4) {                   // K = 0..127 : agg/ws
    v2f a = *(const v2f*)(ap0 + k);
    a.x *= scale; a.y *= scale;
    v2f b = *(const v2f*)(brow + k);
    c = __builtin_amdgcn_wmma_f32_16x16x4_f32(false, a, false, b, (short)0, c,
                                              false, false);
  }
#pragma unroll 8
  for (int k = 0; k < DIM; k += 4) {                   // K = 128..255 : h
    v2f a = *(const v2f*)(ap1 + k);
    v2f b = *(const v2f*)(brow + DIM + k);
    c = __builtin_amdgcn_wmma_f32_16x16x4_f32(false, a, false, b, (short)0, c,
                                              false, false);
  }

  if (threadIdx.x < 16) ssq[threadIdx.x] = 0.0f;
  __syncthreads();

  const int col = col0 + lr;
  const float bias = Wb[col];
  float zv[8];
#pragma unroll
  for (int v = 0; v < 8; ++v) {
    float x = fmaxf(c[v] + bias, 0.0f);
    zv[v] = x;
    atomicAdd(&ssq[v + 8 * hi], x * x);                // ds_add_f32
  }
  __syncthreads();

#pragma unroll
  for (int v = 0; v < 8; ++v) {
    const int r = v + 8 * hi;
    float nrm = sqrtf(ssq[r]);
    nrm = (nrm == 0.0f) ? 1.0f : nrm;
    hout[(size_t)(row0 + r) * DIM + col] = zv[v] / nrm;
  }
}

// ---------------------------------------------------------------------------
extern "C" void kernel_launch(void* const* d_in, const int* in_sizes, int n_in,
                              void* d_out, int out_size, void* d_ws, size_t ws_size,
                              hipStream_t stream) {
  const float* h0   = (const float*)d_in[0];   // [N,128]
  const float* ew   = (const float*)d_in[1];   // [2,E]
  const float* Qw   = (const float*)d_in[2];   // [2,128,128]
  const float* Qb   = (const float*)d_in[3];   // [2,128]
  const float* Ww   = (const float*)d_in[4];   // [2,128,256]
  const float* Wb   = (const float*)d_in[5];   // [2,128]
  const int*   esrc = (const int*)d_in[6];     // [2,E]
  const int*   edst = (const int*)d_in[7];     // [2,E]
  float* out = (float*)d_out;                  // [N,128]

  // workspace layout (floats): agg[N*128] | wsum[N] | nbuf[N*128]  (~51.5 MB)
  float* agg  = (float*)d_ws;
  float* wsum = agg + (size_t)N_NODES * DIM;
  float* nbuf = wsum + N_NODES;

  const int gemm_blocks = N_NODES / 16;                 // 3125
  const int edge_blocks = (N_EDGES + 7) / 8;            // 8 edges (waves)/block
  const size_t zcount = (size_t)N_NODES * DIM + N_NODES;

  for (int l = 0; l < 2; ++l) {
    const float* hin = (l == 0) ? h0 : out;
    zero_f32<<<2048, 256, 0, stream>>>(agg, zcount);
    gemm_q_relu<<<gemm_blocks, 256, 0, stream>>>(
        hin, Qw + (size_t)l * DIM * DIM, Qb + (size_t)l * DIM, nbuf);
    edge_scatter<<<edge_blocks, 256, 0, stream>>>(
        nbuf, ew + (size_t)l * N_EDGES, esrc + (size_t)l * N_EDGES,
        edst + (size_t)l * N_EDGES, agg, wsum);
    gemm_w_relu_norm<<<gemm_blocks, 256, 0, stream>>>(
        agg, wsum, hin, Ww + (size_t)l * DIM * 2 * DIM, Wb + (size_t)l * DIM,
        out);
  }
}